// LogSignature_51573967290391
// MI455X (gfx1250) — compile-verified
//
#include <hip/hip_runtime.h>

// ---------------------------------------------------------------------------
// Depth-4 log-signature of paths [B, 512, 8] -> [B, 4680]  (f32)
//
// Strategy (MI455X / gfx1250, wave32):
//  * Chen product is associative -> split the 511-increment scan into 4
//    chunks, one wave32 each (4 waves / block, 1 block / batch).
//  * Per-step rank-1 Chen update kept fully register-resident: lane L owns
//    the slab i=L/4, j in {2(L%4), 2(L%4)+1} of every tensor level
//    (1 + 2 + 16 + 128 floats/lane), x is wave-uniform.
//  * Cross-chunk Chen combine and tensor-log level-4 terms are genuine
//    64x8x64 matmuls on the 64x64 matrix view of level 4:
//        a1 (x) b3 = kron(a1,I8) @ B3,   a3 (x) b1 = A3 @ kron(I8,b1),
//        a2 (x) b2 = rank-1 outer
//    -> 5 x v_wmma_f32_16x16x4_f32 per 16x16 tile, f32 accumulate.
//  * All conditional fragment elements are built with branch-free selects
//    (loads issued unconditionally) so EXEC stays all-ones around WMMA.
// ---------------------------------------------------------------------------

static constexpr int PLEN   = 512;                  // L
static constexpr int PCH    = 8;                    // C
static constexpr int SIG_SZ = 8 + 64 + 512 + 4096;  // 4680
static constexpr int OFF1 = 0, OFF2 = 8, OFF3 = 72, OFF4 = 584;
static constexpr int NWAVE = 4;

typedef float v2f __attribute__((ext_vector_type(2)));
typedef float v8f __attribute__((ext_vector_type(8)));

__device__ __forceinline__ v8f wmma_f32_16x16x4(v2f a, v2f b, v8f c) {
  // (neg_a, A, neg_b, B, c_mod, C, reuse_a, reuse_b)
  return __builtin_amdgcn_wmma_f32_16x16x4_f32(false, a, false, b, (short)0, c,
                                               false, false);
}

// Level-4 Chen combine for one 16x16 tile of the 64x64 view, in place on S0:
//   c4 = a4 + b4 + a1(x)b3 + a2(x)b2 + a3(x)b1
__device__ __forceinline__ void chen4_tile(float* S0, const float* Sb,
                                           int tp, int tq, int lane) {
  const int lh = lane & 15;
  const int hi = lane >> 4;
  const int p  = tp * 16 + lh;   // A row
  const int q  = tq * 16 + lh;   // B / D column
  const int ip = p >> 3, jp = p & 7;
  const int qh = q >> 3, ql = q & 7;

  v8f c;
#pragma unroll
  for (int r = 0; r < 8; ++r) {
    const int row = tp * 16 + 8 * hi + r;
    c[r] = S0[OFF4 + row * 64 + q] + Sb[OFF4 + row * 64 + q];
  }

  const float a1v = S0[OFF1 + ip];
  const float b1v = Sb[OFF1 + ql];
  const float a2p = S0[OFF2 + p];   // unconditional loads, selected below
  const float b2q = Sb[OFF2 + q];

#pragma unroll
  for (int c0 = 0; c0 < 8; c0 += 4) {
    const int k0 = c0 + 2 * hi;
    v2f a, bb;
    // a1 (x) b3 : A = kron(a1, I8), B = b3 viewed as [8 x 64]
    a.x  = (jp == k0    ) ? a1v : 0.f;
    a.y  = (jp == k0 + 1) ? a1v : 0.f;
    bb.x = Sb[OFF3 + (k0    ) * 64 + q];
    bb.y = Sb[OFF3 + (k0 + 1) * 64 + q];
    c = wmma_f32_16x16x4(a, bb, c);
    // a3 (x) b1 : A = a3 viewed as [64 x 8], B = kron(I8, b1)
    a.x  = S0[OFF3 + p * 8 + k0];
    a.y  = S0[OFF3 + p * 8 + k0 + 1];
    bb.x = (qh == k0    ) ? b1v : 0.f;
    bb.y = (qh == k0 + 1) ? b1v : 0.f;
    c = wmma_f32_16x16x4(a, bb, c);
  }
  // a2 (x) b2 : rank-1 outer product in K slot 0 (branch-free selects)
  {
    v2f a, bb;
    a.x  = (hi == 0) ? a2p : 0.f;
    a.y  = 0.f;
    bb.x = (hi == 0) ? b2q : 0.f;
    bb.y = 0.f;
    c = wmma_f32_16x16x4(a, bb, c);
  }
#pragma unroll
  for (int r = 0; r < 8; ++r) {
    const int row = tp * 16 + 8 * hi + r;
    S0[OFF4 + row * 64 + q] = c[r];
  }
}

// Level-4 of log(S): s4 - (s1 x s3 + s2 x s2 + s3 x s1)/2
//                       + (p22 x s2 + p23 x s1)/3 - (p33 x s1)/4
__device__ __forceinline__ void log4_tile(const float* S, const float* P22,
                                          const float* P23, const float* P33,
                                          float* outb, int tp, int tq, int lane) {
  const int lh = lane & 15;
  const int hi = lane >> 4;
  const int p  = tp * 16 + lh;
  const int q  = tq * 16 + lh;
  const int ip = p >> 3, jp = p & 7;
  const int qh = q >> 3, ql = q & 7;

  v8f c;
#pragma unroll
  for (int r = 0; r < 8; ++r) {
    const int row = tp * 16 + 8 * hi + r;
    c[r] = S[OFF4 + row * 64 + q];
  }

  const float a1v = -0.5f * S[OFF1 + ip];
  const float s1q = S[OFF1 + ql];
  const float uP  = -0.5f * S[OFF2 + p] + (1.f / 3.f) * P22[p];  // unconditional
  const float s2q = S[OFF2 + q];

#pragma unroll
  for (int c0 = 0; c0 < 8; c0 += 4) {
    const int k0 = c0 + 2 * hi;
    v2f a, bb;
    // -(1/2) s1 (x) s3
    a.x  = (jp == k0    ) ? a1v : 0.f;
    a.y  = (jp == k0 + 1) ? a1v : 0.f;
    bb.x = S[OFF3 + (k0    ) * 64 + q];
    bb.y = S[OFF3 + (k0 + 1) * 64 + q];
    c = wmma_f32_16x16x4(a, bb, c);
    // (-(1/2) s3 + (1/3) p23 - (1/4) p33) (x) s1
    a.x = -0.5f * S[OFF3 + p * 8 + k0] + (1.f / 3.f) * P23[p * 8 + k0]
          - 0.25f * P33[p * 8 + k0];
    a.y = -0.5f * S[OFF3 + p * 8 + k0 + 1] + (1.f / 3.f) * P23[p * 8 + k0 + 1]
          - 0.25f * P33[p * 8 + k0 + 1];
    bb.x = (qh == k0    ) ? s1q : 0.f;
    bb.y = (qh == k0 + 1) ? s1q : 0.f;
    c = wmma_f32_16x16x4(a, bb, c);
  }
  // (-(1/2) s2 + (1/3) p22) (x) s2 : rank-1, branch-free
  {
    v2f a, bb;
    a.x  = (hi == 0) ? uP : 0.f;
    a.y  = 0.f;
    bb.x = (hi == 0) ? s2q : 0.f;
    bb.y = 0.f;
    c = wmma_f32_16x16x4(a, bb, c);
  }
#pragma unroll
  for (int r = 0; r < 8; ++r) {
    const int row = tp * 16 + 8 * hi + r;
    outb[OFF4 + row * 64 + q] = c[r];
  }
}

__global__ __launch_bounds__(128, 1) void logsig_kernel(
    const float* __restrict__ path, float* __restrict__ out) {
  __shared__ float smem[NWAVE * SIG_SZ];
  const int tid  = threadIdx.x;
  const int lane = tid & 31;
  const int w    = tid >> 5;
  const int b    = blockIdx.x;

  // ------------- Phase 1: per-wave serial Chen scan over one chunk ---------
  const int i  = lane >> 2;   // lane's level-1 index
  const int jj = lane & 3;    // lane's j-pair

  float a1 = 0.f;
  float a2[2] = {0.f, 0.f};
  float a3[16];
  float a4[128];
#pragma unroll
  for (int n = 0; n < 16; ++n) a3[n] = 0.f;
#pragma unroll
  for (int n = 0; n < 128; ++n) a4[n] = 0.f;

  const int nInc  = PLEN - 1;                        // 511
  const int chunk = (nInc + NWAVE - 1) / NWAVE;      // 128
  const int cs    = w * chunk;
  const int ce    = min(nInc, cs + chunk);

  const float* prow = path + ((size_t)b * PLEN + cs) * PCH;
  float4 r0 = ((const float4*)prow)[0];
  float4 r1 = ((const float4*)prow)[1];

  for (int t = cs; t < ce; ++t) {
    const float* nrow = path + ((size_t)b * PLEN + t + 1) * PCH;
    const float4 n0 = ((const float4*)nrow)[0];
    const float4 n1 = ((const float4*)nrow)[1];
    float x[8];
    x[0] = n0.x - r0.x; x[1] = n0.y - r0.y; x[2] = n0.z - r0.z; x[3] = n0.w - r0.w;
    x[4] = n1.x - r1.x; x[5] = n1.y - r1.y; x[6] = n1.z - r1.z; x[7] = n1.w - r1.w;
    r0 = n0; r1 = n1;

    const float xi = x[i];
#pragma unroll
    for (int j = 0; j < 2; ++j) {
      const float xJ = x[2 * jj + j];
      // level 4 (uses old a1,a2,a3)
      const float t1 = (xi * (1.f / 24.f) + a1 * (1.f / 6.f)) * xJ + 0.5f * a2[j];
#pragma unroll
      for (int k = 0; k < 8; ++k) {
        const float t2 = t1 * x[k] + a3[j * 8 + k];
#pragma unroll
        for (int l = 0; l < 8; ++l) a4[(j * 8 + k) * 8 + l] += t2 * x[l];
      }
      // level 3 (uses old a1,a2)
      const float u3 = (xi * (1.f / 6.f) + 0.5f * a1) * xJ + a2[j];
#pragma unroll
      for (int k = 0; k < 8; ++k) a3[j * 8 + k] += u3 * x[k];
      // level 2 (uses old a1)
      a2[j] += (0.5f * xi + a1) * xJ;
    }
    a1 += xi;
  }

  // store partial signature to LDS slot w (lane-slab -> flat layout).
  // a3/a4 slabs are 16B-aligned and contiguous -> float4 (ds_store_b128).
  {
    float* S = smem + w * SIG_SZ;
    if (jj == 0) S[OFF1 + i] = a1;
#pragma unroll
    for (int j = 0; j < 2; ++j) {
      const int J = 2 * jj + j;
      S[OFF2 + i * 8 + J] = a2[j];
      {
        float4* d3 = (float4*)&S[OFF3 + i * 64 + J * 8];
        d3[0] = make_float4(a3[j * 8 + 0], a3[j * 8 + 1], a3[j * 8 + 2], a3[j * 8 + 3]);
        d3[1] = make_float4(a3[j * 8 + 4], a3[j * 8 + 5], a3[j * 8 + 6], a3[j * 8 + 7]);
      }
#pragma unroll
      for (int k = 0; k < 8; ++k) {
        float4* d4 = (float4*)&S[OFF4 + i * 512 + J * 64 + k * 8];
        const int base = (j * 8 + k) * 8;
        d4[0] = make_float4(a4[base + 0], a4[base + 1], a4[base + 2], a4[base + 3]);
        d4[1] = make_float4(a4[base + 4], a4[base + 5], a4[base + 6], a4[base + 7]);
      }
    }
  }
  __syncthreads();

  // ------------- Phase 2: fold the 4 partial signatures (Chen), in place ---
  float* S0 = smem;
  for (int m = 1; m < NWAVE; ++m) {
    const float* Sb = smem + m * SIG_SZ;
    // level 4: 16 tiles of the 64x64 view, 4 tiles per wave, WMMA
#pragma unroll
    for (int tt = 0; tt < 4; ++tt) {
      const int tile = w * 4 + tt;
      chen4_tile(S0, Sb, tile >> 2, tile & 3, lane);
    }
    __syncthreads();
    // level 3: c3 = a3 + b3 + a1(x)b2 + a2(x)b1
    for (int e = tid; e < 512; e += 128) {
      S0[OFF3 + e] = S0[OFF3 + e] + Sb[OFF3 + e]
                   + S0[OFF1 + (e >> 6)] * Sb[OFF2 + (e & 63)]
                   + S0[OFF2 + (e >> 3)] * Sb[OFF1 + (e & 7)];
    }
    __syncthreads();
    // level 2: c2 = a2 + b2 + a1(x)b1
    if (tid < 64)
      S0[OFF2 + tid] = S0[OFF2 + tid] + Sb[OFF2 + tid]
                     + S0[OFF1 + (tid >> 3)] * Sb[OFF1 + (tid & 7)];
    __syncthreads();
    // level 1
    if (tid < 8) S0[OFF1 + tid] += Sb[OFF1 + tid];
    __syncthreads();
  }

  // ------------- Phase 3: tensor log -> output -----------------------------
  float* P22 = smem + SIG_SZ;        // s1 (x) s1                 [64]
  float* P23 = smem + SIG_SZ + 64;   // s1 (x) s2 + s2 (x) s1     [512]
  float* P33 = smem + SIG_SZ + 576;  // s1 (x) s1 (x) s1          [512]
  if (tid < 64) P22[tid] = S0[tid >> 3] * S0[tid & 7];
  for (int e = tid; e < 512; e += 128) {
    const float s1i = S0[e >> 6], s1j = S0[(e >> 3) & 7], s1k = S0[e & 7];
    P23[e] = s1i * S0[OFF2 + (e & 63)] + S0[OFF2 + (e >> 3)] * s1k;
    P33[e] = s1i * s1j * s1k;
  }
  __syncthreads();

  const size_t obase = (size_t)b * SIG_SZ;
  // level 4 via WMMA
#pragma unroll
  for (int tt = 0; tt < 4; ++tt) {
    const int tile = w * 4 + tt;
    log4_tile(S0, P22, P23, P33, out + obase, tile >> 2, tile & 3, lane);
  }
  // levels 1..3
  if (tid < 8)  out[obase + OFF1 + tid] = S0[OFF1 + tid];
  if (tid < 64) out[obase + OFF2 + tid] = S0[OFF2 + tid] - 0.5f * P22[tid];
  for (int e = tid; e < 512; e += 128)
    out[obase + OFF3 + e] =
        S0[OFF3 + e] - 0.5f * P23[e] + (1.f / 3.f) * P33[e];
}

extern "C" void kernel_launch(void* const* d_in, const int* in_sizes, int n_in,
                              void* d_out, int out_size, void* d_ws,
                              size_t ws_size, hipStream_t stream) {
  (void)n_in; (void)out_size; (void)d_ws; (void)ws_size;
  const float* path = (const float*)d_in[0];
  float* out = (float*)d_out;
  const int B = in_sizes[0] / (PLEN * PCH);   // 256
  logsig_kernel<<<dim3(B), dim3(NWAVE * 32), 0, stream>>>(path, out);
}